// SmallGNN_51135880626276
// MI455X (gfx1250) — compile-verified
//
#include <hip/hip_runtime.h>

// GraphSAGE 3-layer forward for MI455X (gfx1250).
// Phase 1 (per layer): edge scatter-add aggregation (L2-resident, atomic f32).
// Phase 2 (per layer): fused dual-GEMM  out = ReLU(mean@Wl^T + x@Wr^T + b)
//                      using V_WMMA_F32_16X16X4_F32 (full fp32 matrix path).

#define N_NODES 100000
#define E_EDGES 1600000
#define DOUT    256

typedef __attribute__((ext_vector_type(2))) float v2f;
typedef __attribute__((ext_vector_type(8))) float v8f;

// ---------------------------------------------------------------- utilities
__global__ void gnn_zero_kernel(float* __restrict__ p, size_t n) {
    size_t i = (size_t)blockIdx.x * blockDim.x + threadIdx.x;
    size_t stride = (size_t)gridDim.x * blockDim.x;
    for (; i < n; i += stride) p[i] = 0.0f;
}

__global__ void gnn_deg_kernel(const int* __restrict__ dst, float* __restrict__ deg) {
    int e = blockIdx.x * blockDim.x + threadIdx.x;
    if (e < E_EDGES) atomicAdd(&deg[dst[e]], 1.0f);
}

__global__ void gnn_deg_inv_kernel(float* __restrict__ deg) {
    int i = blockIdx.x * blockDim.x + threadIdx.x;
    if (i < N_NODES) {
        float d = deg[i];
        deg[i] = (d > 0.0f) ? (1.0f / d) : 0.0f;
    }
}

// ------------------------------------------------------------- aggregation
// One wave32 per edge; lanes cooperatively gather src row (float4) and
// scatter-add into agg[dst]. x and agg are both L2-resident (~100 MB each).
template <int DIN>
__global__ __launch_bounds__(256)
void gnn_aggregate_kernel(const float* __restrict__ x,
                          const int* __restrict__ src,
                          const int* __restrict__ dst,
                          float* __restrict__ agg) {
    const int lane = threadIdx.x & 31;
    const int e = blockIdx.x * (blockDim.x >> 5) + (threadIdx.x >> 5);
    if (e >= E_EDGES) return;
    const int s = src[e];
    const int d = dst[e];
    const float4* xs = (const float4*)(x + (size_t)s * DIN);
    float* ad = agg + (size_t)d * DIN;
#pragma unroll
    for (int c = lane; c < DIN / 4; c += 32) {
        float4 v = xs[c];
        atomicAdd(&ad[c * 4 + 0], v.x);
        atomicAdd(&ad[c * 4 + 1], v.y);
        atomicAdd(&ad[c * 4 + 2], v.z);
        atomicAdd(&ad[c * 4 + 3], v.w);
    }
}

// ------------------------------------------------------------- fused GEMM
// Wave computes one 16x16 output tile: acc0 accumulates (deg_inv*agg)@Wl^T,
// acc1 accumulates x@Wr^T (two independent WMMA dependency chains).
// A frag (16x4 f32): lanes 0-15 hold M=0..15 K=k+{0,1}; lanes 16-31 K=k+{2,3}.
// B frag (4x16):     lanes 0-15 hold N=0..15 K=k+{0,1}; lanes 16-31 K=k+{2,3}.
// C/D: VGPR r -> M=r (lanes 0-15) / M=r+8 (lanes 16-31), N = lane&15.
template <int DIN>
__global__ __launch_bounds__(128)
void gnn_sage_gemm_kernel(const float* __restrict__ x,
                          const float* __restrict__ agg,
                          const float* __restrict__ deg_inv,
                          const float* __restrict__ Wl,
                          const float* __restrict__ Wr,
                          const float* __restrict__ bias,
                          float* __restrict__ out) {
    const int lane = threadIdx.x & 31;
    const int wave = threadIdx.x >> 5;
    const int row0 = blockIdx.x * 16;
    const int col0 = blockIdx.y * 64 + wave * 16;
    const int m = lane & 15;       // A-fragment row
    const int half = lane >> 4;    // 0: K+{0,1}, 1: K+{2,3}
    const int n = lane & 15;       // B-fragment column

    const float scale = deg_inv[row0 + m];
    const float* arow  = agg + (size_t)(row0 + m) * DIN + 2 * half;
    const float* xrow  = x   + (size_t)(row0 + m) * DIN + 2 * half;
    const float* wlrow = Wl  + (size_t)(col0 + n) * DIN + 2 * half;
    const float* wrrow = Wr  + (size_t)(col0 + n) * DIN + 2 * half;

    v8f acc0 = {};
    v8f acc1 = {};
#pragma unroll 4
    for (int k = 0; k < DIN; k += 4) {
        v2f a0 = *(const v2f*)(arow + k);
        a0 *= scale;                                   // fold mean = agg*deg_inv
        v2f b0 = *(const v2f*)(wlrow + k);
        acc0 = __builtin_amdgcn_wmma_f32_16x16x4_f32(
            false, a0, false, b0, (short)0, acc0, false, false);

        v2f a1 = *(const v2f*)(xrow + k);
        v2f b1 = *(const v2f*)(wrrow + k);
        acc1 = __builtin_amdgcn_wmma_f32_16x16x4_f32(
            false, a1, false, b1, (short)0, acc1, false, false);
    }

    const float bn = bias[col0 + n];
#pragma unroll
    for (int r = 0; r < 8; ++r) {
        const int mm = r + 8 * half;
        float v = acc0[r] + acc1[r] + bn;
        out[(size_t)(row0 + mm) * DOUT + col0 + n] = (v > 0.0f) ? v : 0.0f;
    }
}

// ---------------------------------------------------------------- launcher
extern "C" void kernel_launch(void* const* d_in, const int* in_sizes, int n_in,
                              void* d_out, int out_size, void* d_ws, size_t ws_size,
                              hipStream_t stream) {
    (void)in_sizes; (void)n_in; (void)out_size; (void)ws_size;

    const float* feature = (const float*)d_in[0];
    const int*   eidx    = (const int*)d_in[1];   // [2, E]: row0 = src, row1 = dst
    const int*   src     = eidx;
    const int*   dst     = eidx + E_EDGES;
    const float* Wl0 = (const float*)d_in[2];
    const float* Wr0 = (const float*)d_in[3];
    const float* b0  = (const float*)d_in[4];
    const float* Wl1 = (const float*)d_in[5];
    const float* Wr1 = (const float*)d_in[6];
    const float* b1  = (const float*)d_in[7];
    const float* Wl2 = (const float*)d_in[8];
    const float* Wr2 = (const float*)d_in[9];
    const float* b2  = (const float*)d_in[10];

    // Workspace layout (floats): deg_inv | agg | x1 | x2
    float* ws  = (float*)d_ws;
    float* deg = ws;                              // N (padded to 102400)
    float* agg = ws + 102400;                     // N*256
    float* x1  = agg + (size_t)N_NODES * 256;     // N*256
    float* x2  = x1  + (size_t)N_NODES * 256;     // N*256

    // ---- degrees (ws is poisoned; zero every call — deterministic) ----
    gnn_zero_kernel<<<512, 256, 0, stream>>>(deg, (size_t)N_NODES);
    gnn_deg_kernel<<<(E_EDGES + 255) / 256, 256, 0, stream>>>(dst, deg);
    gnn_deg_inv_kernel<<<(N_NODES + 255) / 256, 256, 0, stream>>>(deg);

    const dim3 gemm_grid(N_NODES / 16, DOUT / 64);   // 6250 x 4
    const int  agg_blocks = E_EDGES / 8;             // 8 waves (edges) per block

    // ---- layer 0: din = 128 ----
    gnn_zero_kernel<<<4096, 256, 0, stream>>>(agg, (size_t)N_NODES * 128);
    gnn_aggregate_kernel<128><<<agg_blocks, 256, 0, stream>>>(feature, src, dst, agg);
    gnn_sage_gemm_kernel<128><<<gemm_grid, 128, 0, stream>>>(feature, agg, deg,
                                                             Wl0, Wr0, b0, x1);

    // ---- layer 1: din = 256 ----
    gnn_zero_kernel<<<4096, 256, 0, stream>>>(agg, (size_t)N_NODES * 256);
    gnn_aggregate_kernel<256><<<agg_blocks, 256, 0, stream>>>(x1, src, dst, agg);
    gnn_sage_gemm_kernel<256><<<gemm_grid, 128, 0, stream>>>(x1, agg, deg,
                                                             Wl1, Wr1, b1, x2);

    // ---- layer 2: din = 256, write final output ----
    gnn_zero_kernel<<<4096, 256, 0, stream>>>(agg, (size_t)N_NODES * 256);
    gnn_aggregate_kernel<256><<<agg_blocks, 256, 0, stream>>>(x2, src, dst, agg);
    gnn_sage_gemm_kernel<256><<<gemm_grid, 128, 0, stream>>>(x2, agg, deg,
                                                             Wl2, Wr2, b2, (float*)d_out);
}